// PairManifoldConstrainedHyperConnections_13168369729882
// MI455X (gfx1250) — compile-verified
//
#include <hip/hip_runtime.h>
#include <math.h>

// ---------------------------------------------------------------------------
// Fused RMSNorm + (x @ [W_pre;W_post;W_res]^T) + tanh + sigmoid/sinkhorn.
// Memory-bound: 134 MB of x streamed once (~6 us floor at 23.3 TB/s).
// Matmul uses V_WMMA_F32_16X16X4_F32 (fp32, matches reference precision).
// RMSNorm folded into a per-row scalar applied to the 24 dot products.
// ---------------------------------------------------------------------------

typedef __attribute__((ext_vector_type(2))) float v2f;
typedef __attribute__((ext_vector_type(8))) float v8f;

#define KDIM    512          // c_hidden
#define KSTEPS  (KDIM / 4)   // 128 WMMA k-steps (K=4 each)
#define NF      24           // 4 pre + 4 post + 16 res
#define WSTRIDE 514          // padded LDS row stride (floats) -> bank-friendly
#define WAVES   8            // waves per block
#define TROWS   16           // rows per wave tile
#define EPS     1.1920929e-07f

__global__ __launch_bounds__(WAVES * 32)
void hyperconn_fused(const float* __restrict__ x,
                     const float* __restrict__ Wpre,
                     const float* __restrict__ Wpost,
                     const float* __restrict__ Wres,
                     const float* __restrict__ bpre,
                     const float* __restrict__ bpost,
                     const float* __restrict__ bres,
                     const float* __restrict__ apre_p,
                     const float* __restrict__ apost_p,
                     const float* __restrict__ ares_p,
                     float* __restrict__ out,
                     int nTiles)
{
    // 25 rows: 24 weight rows + 1 zero row for the padded N-tile columns
    __shared__ float sW[25 * WSTRIDE];                // 51,400 B
    __shared__ float sDots[WAVES][TROWS][NF];         // 12,288 B

    const int tid = threadIdx.x;

    // ---- stage concatenated weights into LDS as [feature][k] -------------
    for (int idx = tid; idx < NF * KDIM; idx += WAVES * 32) {
        const int n = idx >> 9;            // feature row 0..23
        const int k = idx & (KDIM - 1);
        float w;
        if (n < 4)      w = Wpre [(size_t)n * KDIM + k];
        else if (n < 8) w = Wpost[(size_t)(n - 4) * KDIM + k];
        else            w = Wres [(size_t)(n - 8) * KDIM + k];
        sW[n * WSTRIDE + k] = w;
    }
    for (int k = tid; k < KDIM; k += WAVES * 32) sW[24 * WSTRIDE + k] = 0.0f;
    __syncthreads();

    const int wave = tid >> 5;
    const int lane = tid & 31;
    const int half = lane >> 4;            // 0: K%4 in {0,1}   1: K%4 in {2,3}
    const int lr   = lane & 15;
    const int koff = half * 2;

    int tileRaw = blockIdx.x * WAVES + wave;
    const bool live = tileRaw < nTiles;
    const int  tile = live ? tileRaw : (nTiles - 1);   // clamp: compute, don't store

    // A-fragment pointer: lane lr holds row lr of the tile, K-offset koff
    const float* aptr = x + ((size_t)tile * TROWS + lr) * KDIM + koff;
    // B-fragment pointers: column n = lane lr (tile0: n=lr, tile1: n=16+lr or zero row)
    const float* bp0 = &sW[lr * WSTRIDE + koff];
    const int n1 = (16 + lr < NF) ? (16 + lr) : 24;    // 24 = zero row
    const float* bp1 = &sW[n1 * WSTRIDE + koff];

    v8f acc0 = {0.f,0.f,0.f,0.f,0.f,0.f,0.f,0.f};
    v8f acc1 = {0.f,0.f,0.f,0.f,0.f,0.f,0.f,0.f};
    float ss = 0.0f;                        // partial sum of squares

#pragma unroll 8
    for (int kk = 0; kk < KSTEPS; ++kk) {
        const v2f a  = *(const v2f*)(aptr + kk * 4);   // global_load_b64
        const v2f b0 = *(const v2f*)(bp0  + kk * 4);   // ds_load_b64
        const v2f b1 = *(const v2f*)(bp1  + kk * 4);   // ds_load_b64
        ss = fmaf(a.x, a.x, ss);
        ss = fmaf(a.y, a.y, ss);
        acc0 = __builtin_amdgcn_wmma_f32_16x16x4_f32(false, a, false, b0,
                                                     (short)0, acc0, false, false);
        acc1 = __builtin_amdgcn_wmma_f32_16x16x4_f32(false, a, false, b1,
                                                     (short)0, acc1, false, false);
    }

    // combine the two K-halves of the row sum-of-squares (lane lr <-> lr+16)
    ss += __shfl_xor(ss, 16, 32);
    const float srow = rsqrtf(ss * (1.0f / (float)KDIM) + EPS);  // RMS scale for row lr

    // scatter D (16x16 f32 layout: VGPR v -> row v / v+8 in lane halves) to LDS
#pragma unroll
    for (int v = 0; v < 8; ++v) {
        const int row = v + half * 8;
        sDots[wave][row][lr] = acc0[v];
        if (lr < 8) sDots[wave][row][16 + lr] = acc1[v];
    }
    __syncthreads();   // order LDS writes before per-row reads

    if (live && lane < TROWS) {
        const float apre  = apre_p[0];
        const float apost = apost_p[0];
        const float ares  = ares_p[0];

        float t[NF];
#pragma unroll
        for (int j = 0; j < NF; ++j)
            t[j] = tanhf(sDots[wave][lane][j] * srow);

        float o[NF];
#pragma unroll
        for (int j = 0; j < 4; ++j) {
            o[j]     = 1.0f / (1.0f + expf(-(apre  * t[j]     + bpre[j])));
            o[4 + j] = 2.0f / (1.0f + expf(-(apost * t[4 + j] + bpost[j])));
        }

        // Sinkhorn-Knopp on the 4x4, 20 iterations, in registers
        float M[16];
#pragma unroll
        for (int i = 0; i < 16; ++i)
            M[i] = expf(ares * t[8 + i] + bres[i]);
        for (int it = 0; it < 20; ++it) {
#pragma unroll
            for (int i = 0; i < 4; ++i) {
                const float r = 1.0f / (M[i*4] + M[i*4+1] + M[i*4+2] + M[i*4+3]);
                M[i*4] *= r; M[i*4+1] *= r; M[i*4+2] *= r; M[i*4+3] *= r;
            }
#pragma unroll
            for (int j = 0; j < 4; ++j) {
                const float r = 1.0f / (M[j] + M[4+j] + M[8+j] + M[12+j]);
                M[j] *= r; M[4+j] *= r; M[8+j] *= r; M[12+j] *= r;
            }
        }
#pragma unroll
        for (int i = 0; i < 16; ++i) o[8 + i] = M[i];

        float4* dst = (float4*)(out + ((size_t)tile * TROWS + lane) * NF);
#pragma unroll
        for (int q = 0; q < 6; ++q)
            dst[q] = make_float4(o[4*q], o[4*q+1], o[4*q+2], o[4*q+3]);
    }
}

extern "C" void kernel_launch(void* const* d_in, const int* in_sizes, int n_in,
                              void* d_out, int out_size, void* d_ws, size_t ws_size,
                              hipStream_t stream) {
    (void)n_in; (void)out_size; (void)d_ws; (void)ws_size;
    const float* x     = (const float*)d_in[0];
    const float* Wpre  = (const float*)d_in[1];
    const float* Wpost = (const float*)d_in[2];
    const float* Wres  = (const float*)d_in[3];
    const float* bpre  = (const float*)d_in[4];
    const float* bpost = (const float*)d_in[5];
    const float* bres  = (const float*)d_in[6];
    const float* apre  = (const float*)d_in[7];
    const float* apost = (const float*)d_in[8];
    const float* ares  = (const float*)d_in[9];
    float* out = (float*)d_out;

    const int rows   = in_sizes[0] / KDIM;     // 65536 for B=1, L=256
    const int nTiles = rows / TROWS;           // 4096
    const int blocks = (nTiles + WAVES - 1) / WAVES;

    hyperconn_fused<<<blocks, WAVES * 32, 0, stream>>>(
        x, Wpre, Wpost, Wres, bpre, bpost, bres, apre, apost, ares, out, nTiles);
}